// MPS_Layer_27608049779365
// MI455X (gfx1250) — compile-verified
//
#include <hip/hip_runtime.h>
#include <hip/hip_bf16.h>

// MPS layer: ordered product of 784 16x16 fp32 matrices per batch element,
// then contraction with 10 "middle" matrices.
// Two-level tree: kernel1 computes 28-site partial products (28 segments),
// kernel2 combines 28 partials per batch and does the class contraction.
// All matmuls via V_WMMA_F32_16X16X4_F32 (K=16 as 4 accumulated chunks).

#define BATCH      256
#define NUM_SITES  784
#define SPLIT      392
#define NSEG       28
#define SEGLEN     28
#define NCLASS     10

typedef __attribute__((ext_vector_type(2))) float v2f;
typedef __attribute__((ext_vector_type(8))) float v8f;

// ---- one chain step: P <- P @ Mnew  (P in D-layout, Mnew given in B-layout) ----
// D layout (16x16 f32 C/D): vgpr r, lane l<16 -> (row=r,   col=l)
//                                   lane l>=16 -> (row=r+8, col=l-16)
// A layout (16x4 chunk c):  vgpr r, lane l: M=l%16, K=4c + r + 2*(l>=16)
// B layout (4x16 chunk c):  vgpr r, lane l: K=4c + r + 2*(l>=16), N=l%16
__device__ __forceinline__ v8f chain_step(v8f P, const v2f Bm[4],
                                          float* __restrict__ lds,
                                          int col, int hi) {
    // spill P (D layout) to LDS row-major
#pragma unroll
    for (int r = 0; r < 8; ++r)
        lds[(r + hi * 8) * 16 + col] = P[r];
    __builtin_amdgcn_wave_barrier();   // same-wave DS ops are in-order; pin compiler
    // gather A chunks (transpose via LDS)
    v2f A[4];
#pragma unroll
    for (int c = 0; c < 4; ++c) {
#pragma unroll
        for (int r = 0; r < 2; ++r)
            A[c][r] = lds[col * 16 + 4 * c + r + 2 * hi];
    }
    v8f acc = {0.f, 0.f, 0.f, 0.f, 0.f, 0.f, 0.f, 0.f};
#pragma unroll
    for (int c = 0; c < 4; ++c)
        acc = __builtin_amdgcn_wmma_f32_16x16x4_f32(
            /*neg_a=*/false, A[c], /*neg_b=*/false, Bm[c],
            /*c_mod=*/(short)0, acc, /*reuse_a=*/false, /*reuse_b=*/false);
    return acc;
}

// ---------------- kernel 1: per-segment partial products ----------------
// grid: 256 batches * 28 segments = 7168 waves, 8 waves (256 thr) per block.
__global__ void __launch_bounds__(256)
mps_partial(const float* __restrict__ x,     // [256,784,2]
            const float* __restrict__ mw,    // [2,784,16,16]
            float* __restrict__ ws)          // [256,28,16,16]
{
    __shared__ float lds_all[8 * 256];
    const int lane  = threadIdx.x & 31;
    const int wib   = threadIdx.x >> 5;
    const int gwave = blockIdx.x * 8 + wib;
    const int batch = gwave / NSEG;
    const int seg   = gwave % NSEG;
    float* lds = &lds_all[wib * 256];

    const int col = lane & 15;
    const int hi  = lane >> 4;          // 0 = lanes 0-15, 1 = lanes 16-31

    const int s0 = seg * SEGLEN;
    const float* W0 = mw;                       // j = 0 plane
    const float* W1 = mw + NUM_SITES * 256;     // j = 1 plane

    // first site: build directly in D layout
    v8f P;
    {
        const int s    = s0;
        int site = s + SPLIT; if (site >= NUM_SITES) site -= NUM_SITES;
        const float x0 = x[(batch * NUM_SITES + site) * 2 + 0];   // wave-uniform
        const float x1 = x[(batch * NUM_SITES + site) * 2 + 1];
        const float* w0 = W0 + s * 256;
        const float* w1 = W1 + s * 256;
#pragma unroll
        for (int r = 0; r < 8; ++r) {
            const int idx = (r + hi * 8) * 16 + col;
            P[r] = x0 * w0[idx] + x1 * w1[idx];
        }
    }

    // remaining sites: stream fresh matrix in B layout, chain-multiply
    for (int i = 1; i < SEGLEN; ++i) {
        const int s    = s0 + i;
        int site = s + SPLIT; if (site >= NUM_SITES) site -= NUM_SITES;
        const float x0 = x[(batch * NUM_SITES + site) * 2 + 0];
        const float x1 = x[(batch * NUM_SITES + site) * 2 + 1];
        const float* w0 = W0 + s * 256;
        const float* w1 = W1 + s * 256;
        v2f Bm[4];
#pragma unroll
        for (int c = 0; c < 4; ++c) {
#pragma unroll
            for (int r = 0; r < 2; ++r) {
                const int idx = (4 * c + r + 2 * hi) * 16 + col;
                Bm[c][r] = x0 * w0[idx] + x1 * w1[idx];
            }
        }
        P = chain_step(P, Bm, lds, col, hi);
    }

    // write partial product (row-major 16x16) to workspace
    float* o = ws + (batch * NSEG + seg) * 256;
#pragma unroll
    for (int r = 0; r < 8; ++r)
        o[(r + hi * 8) * 16 + col] = P[r];
}

// ---------------- kernel 2: combine segments + class contraction ----------------
// grid: 256 blocks of 1 wave (32 threads).
__global__ void __launch_bounds__(32)
mps_combine(const float* __restrict__ ws,    // [256,28,16,16]
            const float* __restrict__ mid,   // [10,16,16]
            float* __restrict__ out)         // [256,10]
{
    __shared__ float lds[256];
    const int lane  = threadIdx.x & 31;
    const int batch = blockIdx.x;
    const int col = lane & 15;
    const int hi  = lane >> 4;

    const float* base = ws + batch * NSEG * 256;

    // load first partial in D layout
    v8f P;
#pragma unroll
    for (int r = 0; r < 8; ++r)
        P[r] = base[(r + hi * 8) * 16 + col];

    // multiply remaining partials in order
    for (int seg = 1; seg < NSEG; ++seg) {
        const float* Ms = base + seg * 256;
        v2f Bm[4];
#pragma unroll
        for (int c = 0; c < 4; ++c) {
#pragma unroll
            for (int r = 0; r < 2; ++r)
                Bm[c][r] = Ms[(4 * c + r + 2 * hi) * 16 + col];
        }
        P = chain_step(P, Bm, lds, col, hi);
    }

    // out[n,o] = sum_{k,l} P[k,l] * mid[o,l,k]
    float sums[NCLASS];
#pragma unroll
    for (int o = 0; o < NCLASS; ++o) {
        float p = 0.f;
#pragma unroll
        for (int r = 0; r < 8; ++r) {
            const int row = r + hi * 8;
            p += P[r] * mid[o * 256 + col * 16 + row];
        }
        // wave32 all-lane reduction
#pragma unroll
        for (int off = 16; off > 0; off >>= 1)
            p += __shfl_xor(p, off, 32);
        sums[o] = p;
    }
    if (lane == 0) {
#pragma unroll
        for (int o = 0; o < NCLASS; ++o)
            out[batch * NCLASS + o] = sums[o];
    }
}

extern "C" void kernel_launch(void* const* d_in, const int* in_sizes, int n_in,
                              void* d_out, int out_size, void* d_ws, size_t ws_size,
                              hipStream_t stream) {
    const float* x   = (const float*)d_in[0];   // inputs [256,784,2]
    const float* mw  = (const float*)d_in[1];   // matrix_weights [2,784,16,16]
    const float* mid = (const float*)d_in[2];   // middle [10,16,16]
    float* out = (float*)d_out;                 // [256,10]
    float* ws  = (float*)d_ws;                  // needs 256*28*256*4 = 7 MB

    // 256 batches * 28 segments = 7168 waves, 8 waves per 256-thread block
    mps_partial<<<(BATCH * NSEG) / 8, 256, 0, stream>>>(x, mw, ws);
    // one wave per batch element
    mps_combine<<<BATCH, 32, 0, stream>>>(ws, mid, out);
}